// CubeFaceNN_81535659147386
// MI455X (gfx1250) — compile-verified
//
#include <hip/hip_runtime.h>

// CubeFaceNN: out[b,k,d,h,w] = relu(x[b,0,d,h,w] - x_shift_k[b,0,d,h,w]),
// shifts (od,oh,ow): k0=(0,-1,-1) k1=(-1,0,-1) k2=(1,-1,-1) k3=(-1,1,-1)
//                    k4=(-1,-1,0) k5=(-1,-1,1), zero outside the 128^3 volume.
//
// Memory-bound: 403 MB NT stores + ~67 MB reads (input stays in L2; NT store
// hint keeps it there). WMMA inapplicable (elementwise stencil, ~0.4 flop/B).

typedef __attribute__((ext_vector_type(4))) float f4;

namespace {
constexpr int  DIM   = 128;
constexpr long PLANE = 128L * 128L;   // 16384
constexpr long VOL   = 128L * PLANE;  // 2097152
}

__device__ __forceinline__ f4 relu_sub(f4 c, f4 s) {
    f4 r = c - s;
    r.x = fmaxf(r.x, 0.f);
    r.y = fmaxf(r.y, 0.f);
    r.z = fmaxf(r.z, 0.f);
    r.w = fmaxf(r.w, 0.f);
    return r;
}

__global__ __launch_bounds__(256) void cubeface_diff_relu(
    const float* __restrict__ x, float* __restrict__ out) {
  const int b = blockIdx.z;
  const int d = blockIdx.y;
  const int h = (blockIdx.x << 3) + (threadIdx.x >> 5);  // 8 rows per block
  const int w = (threadIdx.x & 31) << 2;                 // 4 floats per thread

  const float* __restrict__ xb = x + (long)b * VOL;

  const f4 zero4 = (f4)0.f;

  // Center value (always in range): aligned 128-bit coalesced load.
  const f4 c = *(const f4*)(xb + (long)d * PLANE + (long)h * DIM + w);

  // gfx1250 prefetch: warm the next h-octet of this plane for the next block.
  if (h + 8 < DIM)
    __builtin_prefetch(xb + (long)d * PLANE + (long)(h + 8) * DIM + w, 0, 0);

  // Neighbor rows, zero-filled outside the volume.
  f4 a4 = zero4; float aL = 0.f;   // (d,   h-1): need w-1..w+2
  f4 b4 = zero4; float bL = 0.f;   // (d-1, h  ): need w-1..w+2
  f4 c4 = zero4; float cL = 0.f;   // (d+1, h-1): need w-1..w+2
  f4 d4 = zero4; float dL = 0.f;   // (d-1, h+1): need w-1..w+2
  f4 e4 = zero4; float eR = 0.f;   // (d-1, h-1): need w..w+4

  if (h - 1 >= 0) {
    const float* r = xb + (long)d * PLANE + (long)(h - 1) * DIM + w;
    a4 = *(const f4*)r;
    if (w > 0) aL = r[-1];
  }
  if (d - 1 >= 0) {
    const float* r = xb + (long)(d - 1) * PLANE + (long)h * DIM + w;
    b4 = *(const f4*)r;
    if (w > 0) bL = r[-1];
  }
  if (d + 1 < DIM && h - 1 >= 0) {
    const float* r = xb + (long)(d + 1) * PLANE + (long)(h - 1) * DIM + w;
    c4 = *(const f4*)r;
    if (w > 0) cL = r[-1];
  }
  if (d - 1 >= 0 && h + 1 < DIM) {
    const float* r = xb + (long)(d - 1) * PLANE + (long)(h + 1) * DIM + w;
    d4 = *(const f4*)r;
    if (w > 0) dL = r[-1];
  }
  if (d - 1 >= 0 && h - 1 >= 0) {
    const float* r = xb + (long)(d - 1) * PLANE + (long)(h - 1) * DIM + w;
    e4 = *(const f4*)r;
    if (w + 4 < DIM) eR = r[4];
  }

  // Assemble the six shifted 4-vectors (w-shift done in registers).
  const f4 s0 = (f4){aL, a4.x, a4.y, a4.z};
  const f4 s1 = (f4){bL, b4.x, b4.y, b4.z};
  const f4 s2 = (f4){cL, c4.x, c4.y, c4.z};
  const f4 s3 = (f4){dL, d4.x, d4.y, d4.z};
  const f4 s4 = e4;
  const f4 s5 = (f4){e4.y, e4.z, e4.w, eR};

  // Non-temporal 128-bit stores: 403 MB of output must not thrash L2,
  // so the 67 MB input stays resident for the 7x neighbor re-reads.
  float* o = out + ((long)b * 6L) * VOL + (long)d * PLANE + (long)h * DIM + w;
  __builtin_nontemporal_store(relu_sub(c, s0), (f4*)(o + 0L * VOL));
  __builtin_nontemporal_store(relu_sub(c, s1), (f4*)(o + 1L * VOL));
  __builtin_nontemporal_store(relu_sub(c, s2), (f4*)(o + 2L * VOL));
  __builtin_nontemporal_store(relu_sub(c, s3), (f4*)(o + 3L * VOL));
  __builtin_nontemporal_store(relu_sub(c, s4), (f4*)(o + 4L * VOL));
  __builtin_nontemporal_store(relu_sub(c, s5), (f4*)(o + 5L * VOL));
}

extern "C" void kernel_launch(void* const* d_in, const int* in_sizes, int n_in,
                              void* d_out, int out_size, void* d_ws, size_t ws_size,
                              hipStream_t stream) {
  (void)in_sizes; (void)n_in; (void)out_size; (void)d_ws; (void)ws_size;
  const float* x = (const float*)d_in[0];
  float* out = (float*)d_out;

  // Grid: x = 16 h-octets, y = 128 depth slices, z = 8 batches.
  dim3 grid(16, 128, 8);
  dim3 block(256);  // 8 waves (wave32) per block; 32 w-groups x 8 h-rows
  cubeface_diff_relu<<<grid, block, 0, stream>>>(x, out);
}